// RelationModule_23450521436770
// MI455X (gfx1250) — compile-verified
//
#include <hip/hip_runtime.h>

// RelationModule on gfx1250, fp32 WMMA + Tensor Data Mover path.
//
// Algebraic refactor: attn@u = psi @ (Phi^T U) - diag(psi.phi) U, turning the
// O(B*N^2*F) pairwise computation into O(B*N*F^2) with a 32x32 per-batch core
// matrix. Bandwidth-bound (~17 MB @ 23.3 TB/s). fp32 WMMA 16x16x4 keeps
// reference precision; X tiles are staged HBM->LDS with TDM (tensor_load_to_lds)
// double-buffered against the WMMA work; outputs staged in LDS for b128 stores.

#define B_ 64
#define N_ 1024
#define F_ 32
#define CH 256     // rows of X staged in LDS per chunk (phase 1)
#define NWAVE 8    // 256 threads / wave32
#define ROWS 128   // rows per block (phase 2)

typedef float v2f __attribute__((ext_vector_type(2)));
typedef float v4f __attribute__((ext_vector_type(4)));
typedef float v8f __attribute__((ext_vector_type(8)));
typedef unsigned int u32x4 __attribute__((ext_vector_type(4)));
typedef int i32x4 __attribute__((ext_vector_type(4)));
typedef int i32x8 __attribute__((ext_vector_type(8)));

__device__ __forceinline__ v8f wmma_f32(v2f a, v2f b, v8f c) {
  // 8 args: (neg_a, A, neg_b, B, c_mod, C, reuse_a, reuse_b)
  return __builtin_amdgcn_wmma_f32_16x16x4_f32(false, a, false, b, (short)0, c,
                                               false, false);
}

// A-matrix 16x4 fp32 tile from row-major src: A[m][k] = src[(m0+m)*stride + k0+k]
// lane l16 -> M, v0/v1 -> K = 2*half, 2*half+1
__device__ __forceinline__ v2f ld_a(const float* s, int stride, int m0, int k0,
                                    int l16, int half) {
  const float* p = s + (m0 + l16) * stride + (k0 + 2 * half);
  v2f a;
  a.x = p[0];
  a.y = p[1];
  return a;
}

// B-matrix 4x16 fp32 tile from row-major src: B[k][n] = src[(k0+k)*stride + n0+n]
// lane l16 -> N, halves split K (also serves as "transposed A" load).
__device__ __forceinline__ v2f ld_b(const float* s, int stride, int k0, int n0,
                                    int l16, int half) {
  const float* p = s + (k0 + 2 * half) * stride + (n0 + l16);
  v2f b;
  b.x = p[0];
  b.y = p[stride];
  return b;
}

// LDS byte offset of a __shared__ object (generic LDS addr low 32 bits == offset)
__device__ __forceinline__ unsigned lds_off(const void* p) {
  return (unsigned)(uintptr_t)p;
}

// One-shot TDM DMA of `nelem` contiguous f32 (nelem <= 65535) from global to
// LDS. D# built per CDNA5 ISA 8.3/8.4: 1-D tile, data_size=4B, no pad/iterate/
// gather, workgroup_mask=0 (not in a cluster). Tracked by TENSORcnt.
// clang-23 toolchain: 6-arg builtin (g0, g1, g2, g3, extra, cpol).
__device__ __forceinline__ void tdm_load_f32(const float* gsrc, unsigned ldsoff,
                                             unsigned nelem) {
  const unsigned long long ga = (unsigned long long)(uintptr_t)gsrc;
  u32x4 g0;
  g0.x = 1u;                                                  // count=1 (valid)
  g0.y = ldsoff;                                              // lds_addr
  g0.z = (unsigned)ga;                                        // global_addr lo
  g0.w = (unsigned)((ga >> 32) & 0x01FFFFFFull) | 0x80000000u;  // hi | type=2
  i32x8 g1;
  g1[0] = 0x00020000;                          // data_size = 2 -> 4 bytes
  g1[1] = (int)((nelem & 0xFFFFu) << 16);      // tensor_dim0 [15:0]
  g1[2] = (int)((nelem >> 16) | (1u << 16));   // tensor_dim0 hi | tensor_dim1=1
  g1[3] = (int)(nelem << 16);                  // tile_dim0 = nelem
  g1[4] = 0;                                   // tile_dim1 = tile_dim2 = 0
  g1[5] = (int)nelem;                          // tensor_dim0_stride lo
  g1[6] = 0;
  g1[7] = 0;
  const i32x4 z4 = {0, 0, 0, 0};
  const i32x8 z8 = {0, 0, 0, 0, 0, 0, 0, 0};
  __builtin_amdgcn_tensor_load_to_lds(g0, g1, z4, z4, z8, 0);
}

// ---------------- Phase 1: per-batch  Mhat = diag(w_psi*w_phi) * (X^T relu(X Wu + bu))
__global__ __launch_bounds__(256) void relmod_phase1(
    const float* __restrict__ x, const float* __restrict__ w_psi,
    const float* __restrict__ w_phi, const float* __restrict__ w_u,
    const float* __restrict__ b_u, float* __restrict__ mhat) {
  __shared__ float Xc[2][CH * F_];       // 2 x 32 KB, TDM double buffer
  __shared__ float Uc[CH * F_];          // 32 KB
  __shared__ float Mp[NWAVE * F_ * F_];  // 32 KB per-wave partials of X^T U
  __shared__ float Wu[F_ * F_];          // 4 KB
  __shared__ float Bu[F_];

  const int b = blockIdx.x;
  const int tid = threadIdx.x;
  const int wave = tid >> 5;
  const int lane = tid & 31;
  const int l16 = lane & 15;
  const int half = lane >> 4;

  const float* xb = x + (size_t)b * (N_ * F_);

  // Kick off DMA of chunk 0 while weights stream in via b128 loads.
  if (wave == 0) tdm_load_f32(xb, lds_off(&Xc[0][0]), CH * F_);
  for (int i = tid; i < (F_ * F_) / 4; i += 256)
    ((v4f*)Wu)[i] = ((const v4f*)w_u)[i];
  if (tid < F_) Bu[tid] = b_u[tid];

  v8f macc[2][2] = {};

  for (int ch = 0; ch < N_ / CH; ++ch) {
    const float* Xcur = &Xc[ch & 1][0];
    if (wave == 0) __builtin_amdgcn_s_wait_tensorcnt(0);
    __syncthreads();  // chunk data in LDS; prior Uc consumers done
    if (wave == 0 && ch + 1 < N_ / CH)
      tdm_load_f32(xb + (size_t)(ch + 1) * (CH * F_),
                   lds_off(&Xc[(ch + 1) & 1][0]), CH * F_);

    // Uc = relu(Xcur @ Wu + Bu); wave owns rows [wave*32, wave*32+32)
#pragma unroll
    for (int s = 0; s < 2; ++s) {
      const int j0 = wave * 32 + s * 16;
#pragma unroll
      for (int gt = 0; gt < 2; ++gt) {
        const int g0 = gt * 16;
        v8f c = {};
#pragma unroll
        for (int k0 = 0; k0 < F_; k0 += 4) {
          v2f a = ld_a(Xcur, F_, j0, k0, l16, half);
          v2f bb = ld_b(Wu, F_, k0, g0, l16, half);
          c = wmma_f32(a, bb, c);
        }
        const float bias = Bu[g0 + l16];
#pragma unroll
        for (int r = 0; r < 8; ++r) {
          float v = c[r] + bias;
          Uc[(j0 + r + 8 * half) * F_ + g0 + l16] = v > 0.0f ? v : 0.0f;
        }
      }
    }
    __syncthreads();

    // macc += Xcur^T @ Uc over this wave's 32-row j slice (K-reduction over j)
#pragma unroll 2
    for (int jj = wave * 32; jj < wave * 32 + 32; jj += 4) {
      v2f at0 = ld_b(Xcur, F_, jj, 0, l16, half);  // A[f][j] = Xcur[j][f]
      v2f at1 = ld_b(Xcur, F_, jj, 16, l16, half);
      v2f bt0 = ld_b(Uc, F_, jj, 0, l16, half);
      v2f bt1 = ld_b(Uc, F_, jj, 16, l16, half);
      macc[0][0] = wmma_f32(at0, bt0, macc[0][0]);
      macc[0][1] = wmma_f32(at0, bt1, macc[0][1]);
      macc[1][0] = wmma_f32(at1, bt0, macc[1][0]);
      macc[1][1] = wmma_f32(at1, bt1, macc[1][1]);
    }
    // top-of-loop barrier protects Uc/Xc reuse
  }

  // stash per-wave partials, reduce, scale rows by w_psi*w_phi
#pragma unroll
  for (int ft = 0; ft < 2; ++ft)
#pragma unroll
    for (int gt = 0; gt < 2; ++gt)
#pragma unroll
      for (int r = 0; r < 8; ++r)
        Mp[wave * (F_ * F_) + (ft * 16 + r + 8 * half) * F_ + gt * 16 + l16] =
            macc[ft][gt][r];
  __syncthreads();

  // 256 lanes x one v4f each: reduce 8 partials, scale row f, b128 store
  {
    const int e4 = tid;  // v4f index within 32x32
    v4f s = {0.0f, 0.0f, 0.0f, 0.0f};
#pragma unroll
    for (int w = 0; w < NWAVE; ++w) s += ((const v4f*)Mp)[w * 256 + e4];
    const int f = (e4 * 4) >> 5;  // row (F_ == 32)
    const float sc = w_psi[f] * w_phi[f];
    ((v4f*)(mhat + b * (F_ * F_)))[e4] = s * sc;
  }
}

// ---------------- Phase 2: out = ((X Mhat - d .* relu(X Wu + bu)) / N) Wr + X
__global__ __launch_bounds__(256) void relmod_phase2(
    const float* __restrict__ x, const float* __restrict__ w_psi,
    const float* __restrict__ w_phi, const float* __restrict__ w_u,
    const float* __restrict__ b_u, const float* __restrict__ w_r,
    const float* __restrict__ mhat, float* __restrict__ out) {
  __shared__ float Xs[ROWS * F_];  // 16 KB (TDM-filled)
  __shared__ float Us[ROWS * F_];  // 16 KB (later reused as output staging)
  __shared__ float Ag[ROWS * F_];  // 16 KB
  __shared__ float Wu[F_ * F_];
  __shared__ float Wr[F_ * F_];
  __shared__ float Mh[F_ * F_];
  __shared__ float Bu[F_];
  __shared__ float Wpw[F_];
  __shared__ float Dl[ROWS];

  const int rt = blockIdx.x;
  const int b = blockIdx.y;
  const int row0 = rt * ROWS;
  const int tid = threadIdx.x;
  const int wave = tid >> 5;
  const int lane = tid & 31;
  const int l16 = lane & 15;
  const int half = lane >> 4;

  const float* xb = x + (size_t)b * (N_ * F_) + (size_t)row0 * F_;

  // TDM the X tile while weights + core matrix stream in via b128 loads.
  if (wave == 0) tdm_load_f32(xb, lds_off(Xs), ROWS * F_);
  {
    const int i = tid;  // 256 v4f = 1024 floats each
    ((v4f*)Wu)[i] = ((const v4f*)w_u)[i];
    ((v4f*)Wr)[i] = ((const v4f*)w_r)[i];
    ((v4f*)Mh)[i] = ((const v4f*)(mhat + b * (F_ * F_)))[i];
  }
  if (tid < F_) {
    Bu[tid] = b_u[tid];
    Wpw[tid] = w_psi[tid] * w_phi[tid];
  }
  if (wave == 0) __builtin_amdgcn_s_wait_tensorcnt(0);
  __syncthreads();

  const int j0 = wave * 16;  // this wave's 16-row strip

  // Us = relu(Xs @ Wu + Bu)
#pragma unroll
  for (int gt = 0; gt < 2; ++gt) {
    const int g0 = gt * 16;
    v8f c = {};
#pragma unroll
    for (int k0 = 0; k0 < F_; k0 += 4) {
      v2f a = ld_a(Xs, F_, j0, k0, l16, half);
      v2f bb = ld_b(Wu, F_, k0, g0, l16, half);
      c = wmma_f32(a, bb, c);
    }
    const float bias = Bu[g0 + l16];
#pragma unroll
    for (int r = 0; r < 8; ++r) {
      float v = c[r] + bias;
      Us[(j0 + r + 8 * half) * F_ + g0 + l16] = v > 0.0f ? v : 0.0f;
    }
  }

  // d_i = sum_f x_if^2 * w_psi[f] * w_phi[f]  (self-term to subtract)
  if (tid < ROWS) {
    float s = 0.0f;
#pragma unroll
    for (int f = 0; f < F_; ++f) {
      const float v = Xs[tid * F_ + f];
      s += v * v * Wpw[f];
    }
    Dl[tid] = s;
  }
  __syncthreads();

  // Ag = (Xs @ Mh - d .* Us) * (1/N)
#pragma unroll
  for (int gt = 0; gt < 2; ++gt) {
    const int g0 = gt * 16;
    v8f c = {};
#pragma unroll
    for (int k0 = 0; k0 < F_; k0 += 4) {
      v2f a = ld_a(Xs, F_, j0, k0, l16, half);
      v2f bb = ld_b(Mh, F_, k0, g0, l16, half);
      c = wmma_f32(a, bb, c);
    }
#pragma unroll
    for (int r = 0; r < 8; ++r) {
      const int row = j0 + r + 8 * half;
      const int g = g0 + l16;
      Ag[row * F_ + g] = (c[r] - Dl[row] * Us[row * F_ + g]) * (1.0f / N_);
    }
  }
  __syncthreads();

  // out = Ag @ Wr + Xs; stage into Us (dead after Ag) for coalesced b128 stores
#pragma unroll
  for (int gt = 0; gt < 2; ++gt) {
    const int g0 = gt * 16;
    v8f c = {};
#pragma unroll
    for (int k0 = 0; k0 < F_; k0 += 4) {
      v2f a = ld_a(Ag, F_, j0, k0, l16, half);
      v2f bb = ld_b(Wr, F_, k0, g0, l16, half);
      c = wmma_f32(a, bb, c);
    }
#pragma unroll
    for (int r = 0; r < 8; ++r) {
      const int row = j0 + r + 8 * half;
      const int g = g0 + l16;
      Us[row * F_ + g] = c[r] + Xs[row * F_ + g];
    }
  }
  __syncthreads();

  float* outp = out + (size_t)b * (N_ * F_) + (size_t)row0 * F_;
#pragma unroll
  for (int i = tid; i < (ROWS * F_) / 4; i += 256)
    ((v4f*)outp)[i] = ((const v4f*)Us)[i];
}

extern "C" void kernel_launch(void* const* d_in, const int* in_sizes, int n_in,
                              void* d_out, int out_size, void* d_ws,
                              size_t ws_size, hipStream_t stream) {
  const float* x = (const float*)d_in[0];
  const float* w_psi = (const float*)d_in[1];
  const float* w_phi = (const float*)d_in[2];
  const float* w_u = (const float*)d_in[3];
  const float* b_u = (const float*)d_in[4];
  const float* w_r = (const float*)d_in[5];
  float* out = (float*)d_out;
  float* mhat = (float*)d_ws;  // B*F*F floats = 256 KB scratch

  relmod_phase1<<<dim3(B_), dim3(256), 0, stream>>>(x, w_psi, w_phi, w_u, b_u,
                                                    mhat);
  relmod_phase2<<<dim3(N_ / ROWS, B_), dim3(256), 0, stream>>>(
      x, w_psi, w_phi, w_u, b_u, w_r, mhat, out);
}